// prompt_model_6571299963173
// MI455X (gfx1250) — compile-verified
//
#include <hip/hip_runtime.h>
#include <hip/hip_bf16.h>

typedef float v2f __attribute__((ext_vector_type(2)));
typedef float v8f __attribute__((ext_vector_type(8)));

#define BB   32
#define SS   1024
#define HH   768
#define DD   200
#define MTOT (BB * SS)          // 32768 rows of the score GEMM
#define NT   13                 // ceil(200/16) N-tiles
#define NP   (NT * 16)          // 208 padded N
#define KC   192                // K-chunk staged in LDS (192*208*4 = 156 KB)

// ---------------------------------------------------------------------------
// Kernel 1: s[b,t] = sum_d v[d] * tanh( hidden[b,t,:] . W1[:,d] + b1[d] )
// f32 WMMA GEMM, M=32768, N=208 (zero-padded), K=768.
// - W1 staged per K-chunk in LDS, tail columns zeroed at staging time
//   -> branch-free WMMA inner loop (no EXEC-mask dance).
// - 13 persistent v8f accumulators per wave -> hidden read exactly once.
// One wave per 16-row M tile; 8 waves / block; 256 blocks.
// ---------------------------------------------------------------------------
__global__ __launch_bounds__(256) void score_gemm_kernel(
    const float* __restrict__ hidden,   // [MTOT, HH]
    const float* __restrict__ W1,       // [HH, DD]
    const float* __restrict__ b1,       // [DD]
    const float* __restrict__ v,        // [DD]
    float* __restrict__ s_out)          // [MTOT]
{
    extern __shared__ float sW[];       // [KC][NP]

    const int tid   = threadIdx.x;
    const int lane  = tid & 31;
    const int wave  = tid >> 5;
    const int m0    = (blockIdx.x * 8 + wave) * 16;

    // A-matrix (16x4 f32) layout: lanes 0-15 -> M=lane, K pair {0,1};
    // lanes 16-31 -> M=lane-16, K pair {2,3}.
    const int row   = lane & 15;
    const int khalf = lane >> 4;
    const int col   = lane & 15;        // N within tile (B/C/D layout)

    const float* __restrict__ arow = hidden + (size_t)(m0 + row) * HH;

    v8f c[NT];
#pragma unroll
    for (int nt = 0; nt < NT; ++nt)
        c[nt] = (v8f){0.f, 0.f, 0.f, 0.f, 0.f, 0.f, 0.f, 0.f};

    for (int kc = 0; kc < HH; kc += KC) {
        __syncthreads();                // protect previous chunk from overwrite
        // Cooperative stage of W1[kc:kc+KC, 0:NP] with zero-padded tail.
        for (int idx = tid; idx < KC * NP; idx += 256) {
            const int kk = idx / NP;
            const int nn = idx - kk * NP;
            sW[idx] = (nn < DD) ? W1[(size_t)(kc + kk) * DD + nn] : 0.0f;
        }
        __syncthreads();

        for (int k = 0; k < KC; k += 4) {
            v2f a;
            a.x = arow[kc + k + 2 * khalf + 0];
            a.y = arow[kc + k + 2 * khalf + 1];

            const float* __restrict__ br0 = &sW[(k + 2 * khalf + 0) * NP + col];
            const float* __restrict__ br1 = &sW[(k + 2 * khalf + 1) * NP + col];
#pragma unroll
            for (int nt = 0; nt < NT; ++nt) {
                v2f bm;
                bm.x = br0[nt * 16];
                bm.y = br1[nt * 16];
                // (neg_a, A, neg_b, B, c_mod, C, reuse_a, reuse_b)
                c[nt] = __builtin_amdgcn_wmma_f32_16x16x4_f32(
                            false, a, false, bm, (short)0, c[nt], false, false);
            }
        }
    }

    // Epilogue: s partial = sum_n v[n] * tanh(c + b1[n]); pad columns get v=0.
    float sacc[8];
#pragma unroll
    for (int r = 0; r < 8; ++r) sacc[r] = 0.0f;

#pragma unroll
    for (int nt = 0; nt < NT; ++nt) {
        const int   n   = nt * 16 + col;
        const bool  nok = (n < DD);
        const float b1n = nok ? b1[n] : 0.0f;
        const float vn  = nok ? v[n]  : 0.0f;
#pragma unroll
        for (int r = 0; r < 8; ++r)
            sacc[r] += vn * tanhf(c[nt][r] + b1n);
    }

    // Reduce over the 16 lanes of each half-wave group; C layout:
    // VGPR r, lanes 0-15 -> M=r ; lanes 16-31 -> M=r+8.
#pragma unroll
    for (int r = 0; r < 8; ++r) {
        float x = sacc[r];
#pragma unroll
        for (int off = 8; off > 0; off >>= 1)
            x += __shfl_xor(x, off, 16);
        sacc[r] = x;
    }
    if (col == 0) {
#pragma unroll
        for (int r = 0; r < 8; ++r)
            s_out[m0 + r + 8 * khalf] = sacc[r];
    }
}

// ---------------------------------------------------------------------------
// Kernel 2: per batch — stable exp and inclusive prefix sum.
// One block of 1024 threads per batch (32 waves).
// ---------------------------------------------------------------------------
__global__ __launch_bounds__(1024) void softmax_scan_kernel(
    const float* __restrict__ s_in,   // [BB, SS]
    float* __restrict__ e_out,        // [BB, SS]
    float* __restrict__ P_out,        // [BB, SS] inclusive prefix of e
    float* __restrict__ Z_out)        // [BB]
{
    __shared__ float buf[SS];
    const int b = blockIdx.x;
    const int t = threadIdx.x;

    const float sv = s_in[b * SS + t];

    // block max
    buf[t] = sv;
    __syncthreads();
    for (int off = SS / 2; off > 0; off >>= 1) {
        if (t < off) buf[t] = fmaxf(buf[t], buf[t + off]);
        __syncthreads();
    }
    const float m = buf[0];
    __syncthreads();

    const float ev = expf(sv - m);
    e_out[b * SS + t] = ev;

    // Hillis-Steele inclusive scan
    buf[t] = ev;
    __syncthreads();
    for (int off = 1; off < SS; off <<= 1) {
        const float add = (t >= off) ? buf[t - off] : 0.0f;
        __syncthreads();
        buf[t] += add;
        __syncthreads();
    }
    P_out[b * SS + t] = buf[t];
    if (t == SS - 1) Z_out[b] = buf[t];
}

// ---------------------------------------------------------------------------
// Kernel 3: bandwidth kernel. grid = (B, H/256), block = 256.
// Thread owns one h column; e/P staged in LDS (broadcast across block).
// prefix  : running scan down k
// postfix : running scan up k
// loo     : total T then (T - e_j*h_j) / ((Z - e_j)*(S-1))
// ---------------------------------------------------------------------------
__global__ __launch_bounds__(256) void attn_apply_kernel(
    const float* __restrict__ hidden,   // [BB, SS, HH]
    const int*   __restrict__ ptypes,   // [BB]
    const float* __restrict__ e_in,     // [BB, SS]
    const float* __restrict__ P_in,     // [BB, SS]
    const float* __restrict__ Z_in,     // [BB]
    float* __restrict__ out)            // [BB, SS, HH]
{
    __shared__ float se[SS];
    __shared__ float sP[SS];

    const int b = blockIdx.x;
    const int h = blockIdx.y * 256 + threadIdx.x;

    for (int i = threadIdx.x; i < SS; i += 256) {
        se[i] = e_in[b * SS + i];
        sP[i] = P_in[b * SS + i];
    }
    __syncthreads();

    const float Zb = Z_in[b];
    const int   pt = ptypes[b];

    const float* __restrict__ hb = hidden + (size_t)b * SS * HH + h;
    float*       __restrict__ ob = out    + (size_t)b * SS * HH + h;

    if (pt == 0) {                      // prefix: k <= j
        float acc = 0.0f;
#pragma unroll 4
        for (int k = 0; k < SS; ++k) {
            __builtin_prefetch(&hb[(size_t)(k + 16) * HH], 0, 0);
            acc += se[k] * hb[(size_t)k * HH];
            const float val = acc / (sP[k] * (float)(k + 1));
            __builtin_nontemporal_store(val, &ob[(size_t)k * HH]);
        }
    } else if (pt == 1) {               // postfix: k >= j
        float acc = 0.0f;
#pragma unroll 4
        for (int k = SS - 1; k >= 0; --k) {
            __builtin_prefetch(&hb[(size_t)(k - 16) * HH], 0, 0);
            acc += se[k] * hb[(size_t)k * HH];
            const float Q = Zb - sP[k] + se[k];     // inclusive suffix sum
            const float val = acc / (Q * (float)(SS - k));
            __builtin_nontemporal_store(val, &ob[(size_t)k * HH]);
        }
    } else {                            // leave-one-out: k != j
        float T = 0.0f;
#pragma unroll 4
        for (int k = 0; k < SS; ++k) {
            __builtin_prefetch(&hb[(size_t)(k + 16) * HH], 0, 0);
            T += se[k] * hb[(size_t)k * HH];
        }
        const float invC = 1.0f / (float)(SS - 1);
#pragma unroll 4
        for (int j = 0; j < SS; ++j) {
            const float x   = se[j] * hb[(size_t)j * HH];
            const float val = (T - x) / (Zb - se[j]) * invC;
            __builtin_nontemporal_store(val, &ob[(size_t)j * HH]);
        }
    }
}

// ---------------------------------------------------------------------------
extern "C" void kernel_launch(void* const* d_in, const int* in_sizes, int n_in,
                              void* d_out, int out_size, void* d_ws, size_t ws_size,
                              hipStream_t stream)
{
    const float* hidden = (const float*)d_in[0];   // [B,S,H]
    const float* W1     = (const float*)d_in[1];   // [H,D]
    const float* b1     = (const float*)d_in[2];   // [D]
    const float* v      = (const float*)d_in[3];   // [D]
    const int*   ptypes = (const int*)  d_in[4];   // [B]
    float*       out    = (float*)d_out;           // [B,S,H]

    float* ws    = (float*)d_ws;
    float* s_buf = ws;               // MTOT floats
    float* e_buf = ws + MTOT;        // MTOT floats
    float* P_buf = ws + 2 * MTOT;    // MTOT floats
    float* Z_buf = ws + 3 * MTOT;    // BB floats

    // 1) scores via f32 WMMA; dynamic LDS = staged W1 K-chunk (156 KB)
    const size_t ldsBytes = (size_t)KC * NP * sizeof(float);
    score_gemm_kernel<<<MTOT / (16 * 8), 256, ldsBytes, stream>>>(
        hidden, W1, b1, v, s_buf);

    // 2) per-batch exp + prefix scan
    softmax_scan_kernel<<<BB, SS, 0, stream>>>(s_buf, e_buf, P_buf, Z_buf);

    // 3) streaming scan application
    dim3 grid3(BB, HH / 256, 1);
    attn_apply_kernel<<<grid3, 256, 0, stream>>>(
        hidden, ptypes, e_buf, P_buf, Z_buf, out);
}